// TGCN_83047487635515
// MI455X (gfx1250) — compile-verified
//
#include <hip/hip_runtime.h>
#include <hip/hip_bf16.h>

// ---------------------------------------------------------------------------
// GraphConv x3 + mean-pool + linear head for MI455X (gfx1250, wave32, WMMA).
//   layer:  out = relu( agg @ W_rel + b + h @ W_root ),  agg = scatter_add(h[src] -> dst)
// GEMMs: native fp32 WMMA (v_wmma_f32_16x16x4_f32), 32-row x 128-col block
// tiles, A tiles staged via global_load_async_to_lds_b128 (ASYNCcnt path).
// ---------------------------------------------------------------------------

#define N_NODES  100000
#define N_EDGES  1600000
#define HID      128
#define OUT_C    10
#define N_GRAPHS 512

typedef float v2f __attribute__((ext_vector_type(2)));
typedef float v8f __attribute__((ext_vector_type(8)));

#define TILE_M    32    // 100000 = 3125 * 32 exactly -> EXEC all-ones everywhere
#define LDS_PITCH 132   // 128 + 4 pad -> stride-128 column reads hit distinct banks

// ---------------- edge scatter-add: agg[dst] += h[src], one wave per edge ---
__global__ __launch_bounds__(256)
void tgcn_scatter_add(const float* __restrict__ h,
                      const int*  __restrict__ src,
                      const int*  __restrict__ dst,
                      float*      __restrict__ agg,
                      int n_edges)
{
    const int wave = (blockIdx.x << 3) + (threadIdx.x >> 5);   // 8 waves / block
    const int lane = threadIdx.x & 31;
    if (wave >= n_edges) return;
    const int s = src[wave];                 // wave-uniform -> scalar load
    const int d = dst[wave];
    const float4 v = *reinterpret_cast<const float4*>(h + (size_t)s * HID + lane * 4);
    float* p = agg + (size_t)d * HID + lane * 4;
    unsafeAtomicAdd(p + 0, v.x);             // native global_atomic_add_f32
    unsafeAtomicAdd(p + 1, v.y);
    unsafeAtomicAdd(p + 2, v.z);
    unsafeAtomicAdd(p + 3, v.w);
}

// ---------------- fused transform: out = [relu]( agg@W_rel + h@W_root + b ) -
// Block = 32 rows x 128 cols, 8 waves; wave owns a 16-col slice with two
// 16x16 M-tiles (two accumulators) so every B fragment feeds two WMMAs.
// In-place safe (out may alias h_in): tiles fully staged to LDS before stores.
__global__ __launch_bounds__(256)
void tgcn_gemm(const float* __restrict__ agg,
               const float* __restrict__ h_in,
               const float* __restrict__ W_rel,   // [128,128] row-major (k, n)
               const float* __restrict__ W_root,  // [128,128] row-major (k, n)
               const float* __restrict__ bias,    // [128]
               float*       __restrict__ h_out,
               int relu)
{
    __shared__ float sAgg[TILE_M][LDS_PITCH];
    __shared__ float sH[TILE_M][LDS_PITCH];

    const int tid  = threadIdx.x;
    const int wave = tid >> 5;          // 0..7
    const int lane = tid & 31;
    const size_t rowBase = (size_t)blockIdx.x * TILE_M;

    // -- async global->LDS staging of both 32x128 tiles (16B chunks) --------
    {
        const float* gA = agg  + rowBase * HID;
        const float* gH = h_in + rowBase * HID;
        #pragma unroll
        for (int j = 0; j < (TILE_M * HID) / (4 * 256); ++j) {  // 4 chunks/thread
            const int ch = tid + j * 256;          // 16-byte chunk id, 0..1023
            const int r  = ch >> 5;                // row within tile
            const int c  = (ch & 31) << 2;         // col (multiple of 4)
            const unsigned ldsA = (unsigned)(uintptr_t)(&sAgg[r][c]); // LDS byte addr
            const unsigned ldsH = (unsigned)(uintptr_t)(&sH[r][c]);
            const unsigned goff = (unsigned)((r * HID + c) * sizeof(float));
            asm volatile("global_load_async_to_lds_b128 %0, %1, %2"
                         :: "v"(ldsA), "v"(goff), "s"(gA) : "memory");
            asm volatile("global_load_async_to_lds_b128 %0, %1, %2"
                         :: "v"(ldsH), "v"(goff), "s"(gH) : "memory");
        }
        asm volatile("s_wait_asynccnt 0x0" ::: "memory");
    }
    __syncthreads();

    const int m  = lane & 15;              // A-matrix row within M-tile
    const int kh = (lane >> 4) << 1;       // K sub-offset: 0 (lanes 0-15) / 2 (16-31)
    const int n  = (wave << 4) + (lane & 15);   // output column

    v8f acc0 = {};   // rows [0,16)
    v8f acc1 = {};   // rows [16,32)
    #pragma unroll 4
    for (int k = 0; k < HID; k += 4) {
        const int kk = k + kh;
        v2f b, a0, a1;
        b.x  = W_rel[(size_t)kk * HID + n];
        b.y  = W_rel[(size_t)(kk + 1) * HID + n];
        a0.x = sAgg[m][kk];
        a0.y = sAgg[m][kk + 1];
        a1.x = sAgg[m + 16][kk];
        a1.y = sAgg[m + 16][kk + 1];
        acc0 = __builtin_amdgcn_wmma_f32_16x16x4_f32(false, a0, false, b,
                                                     (short)0, acc0, false, false);
        acc1 = __builtin_amdgcn_wmma_f32_16x16x4_f32(false, a1, false, b,
                                                     (short)0, acc1, false, false);
    }
    #pragma unroll 4
    for (int k = 0; k < HID; k += 4) {
        const int kk = k + kh;
        v2f b, a0, a1;
        b.x  = W_root[(size_t)kk * HID + n];
        b.y  = W_root[(size_t)(kk + 1) * HID + n];
        a0.x = sH[m][kk];
        a0.y = sH[m][kk + 1];
        a1.x = sH[m + 16][kk];
        a1.y = sH[m + 16][kk + 1];
        acc0 = __builtin_amdgcn_wmma_f32_16x16x4_f32(false, a0, false, b,
                                                     (short)0, acc0, false, false);
        acc1 = __builtin_amdgcn_wmma_f32_16x16x4_f32(false, a1, false, b,
                                                     (short)0, acc1, false, false);
    }

    // D layout: VGPR r -> M = r + 8*(lane>>4), N = lane&15
    const float bv  = bias[n];
    const int   mhi = (lane >> 4) << 3;
    #pragma unroll
    for (int r = 0; r < 8; ++r) {
        float v0 = acc0[r] + bv;
        float v1 = acc1[r] + bv;
        if (relu) { v0 = fmaxf(v0, 0.0f); v1 = fmaxf(v1, 0.0f); }
        h_out[(rowBase + mhi + r) * HID + n]      = v0;
        h_out[(rowBase + 16 + mhi + r) * HID + n] = v1;
    }
}

// ---------------- mean-pool accumulate: wave per node ----------------------
__global__ __launch_bounds__(256)
void tgcn_pool(const float* __restrict__ h,
               const int*  __restrict__ batch,
               float*      __restrict__ pooled,   // [G,128]
               float*      __restrict__ counts,   // [G]
               int n_nodes)
{
    const int node = (blockIdx.x << 3) + (threadIdx.x >> 5);
    const int lane = threadIdx.x & 31;
    if (node >= n_nodes) return;
    const int g = batch[node];
    const float4 v = *reinterpret_cast<const float4*>(h + (size_t)node * HID + lane * 4);
    float* p = pooled + (size_t)g * HID + lane * 4;
    unsafeAtomicAdd(p + 0, v.x);
    unsafeAtomicAdd(p + 1, v.y);
    unsafeAtomicAdd(p + 2, v.z);
    unsafeAtomicAdd(p + 3, v.w);
    if (lane == 0)
        unsafeAtomicAdd(counts + g, 1.0f);
}

// ---------------- head: out = (pooled/count) @ W_lin + b_lin ---------------
__global__ __launch_bounds__(256)
void tgcn_head(const float* __restrict__ pooled,
               const float* __restrict__ counts,
               const float* __restrict__ W_lin,   // [128,10]
               const float* __restrict__ b_lin,   // [10]
               float*       __restrict__ out)     // [G,10]
{
    const int idx = blockIdx.x * blockDim.x + threadIdx.x;
    if (idx >= N_GRAPHS * OUT_C) return;
    const int g = idx / OUT_C;
    const int o = idx % OUT_C;
    const float inv = 1.0f / fmaxf(counts[g], 1.0f);
    float acc = 0.0f;
    #pragma unroll 8
    for (int k = 0; k < HID; ++k)
        acc += pooled[(size_t)g * HID + k] * W_lin[k * OUT_C + o];
    out[idx] = acc * inv + b_lin[o];
}

// ---------------------------------------------------------------------------
extern "C" void kernel_launch(void* const* d_in, const int* in_sizes, int n_in,
                              void* d_out, int out_size, void* d_ws, size_t ws_size,
                              hipStream_t stream) {
    const float* x      = (const float*)d_in[0];
    const int*   ei     = (const int*)d_in[1];
    const int*   batch  = (const int*)d_in[2];
    const float* W_rel[3]  = { (const float*)d_in[3], (const float*)d_in[6], (const float*)d_in[9]  };
    const float* b[3]      = { (const float*)d_in[4], (const float*)d_in[7], (const float*)d_in[10] };
    const float* W_root[3] = { (const float*)d_in[5], (const float*)d_in[8], (const float*)d_in[11] };
    const float* W_lin  = (const float*)d_in[12];
    const float* b_lin  = (const float*)d_in[13];
    float* out = (float*)d_out;

    const int* src = ei;            // edge_index[0]
    const int* dst = ei + N_EDGES;  // edge_index[1]

    // workspace layout
    float* agg    = (float*)d_ws;                              // [N,128]
    float* h      = agg    + (size_t)N_NODES * HID;            // [N,128]
    float* pooled = h      + (size_t)N_NODES * HID;            // [G,128]
    float* counts = pooled + (size_t)N_GRAPHS * HID;           // [G]

    const int scatterBlocks = (N_EDGES + 7) / 8;    // wave per edge, 8 waves/block
    const int gemmBlocks    = N_NODES / TILE_M;     // 100000 = 3125 * 32 exactly
    const int poolBlocks    = (N_NODES + 7) / 8;

    for (int l = 0; l < 3; ++l) {
        const float* hin = (l == 0) ? x : h;
        hipMemsetAsync(agg, 0, (size_t)N_NODES * HID * sizeof(float), stream);
        tgcn_scatter_add<<<scatterBlocks, 256, 0, stream>>>(hin, src, dst, agg, N_EDGES);
        tgcn_gemm<<<gemmBlocks, 256, 0, stream>>>(agg, hin, W_rel[l], W_root[l], b[l],
                                                  h, (l < 2) ? 1 : 0);
    }

    hipMemsetAsync(pooled, 0, ((size_t)N_GRAPHS * HID + N_GRAPHS) * sizeof(float), stream);
    tgcn_pool<<<poolBlocks, 256, 0, stream>>>(h, batch, pooled, counts, N_NODES);
    tgcn_head<<<(N_GRAPHS * OUT_C + 255) / 256, 256, 0, stream>>>(pooled, counts, W_lin, b_lin, out);
}